// FoveateImage_2697239462678
// MI455X (gfx1250) — compile-verified
//
#include <hip/hip_runtime.h>
#include <cstdint>

// ---------------------------------------------------------------------------
// Foveated image rendering, MI455X (gfx1250, wave32).
// Memory-bound fp32 pipeline: fused separable filters, TDM async tile loads to
// LDS (TENSORcnt-pipelined), and the full-res vertical polyphase filter run as
// a banded GEMM on the matrix core via V_WMMA_F32_16X16X4_F32.
// ---------------------------------------------------------------------------

#define SIGMA_F 0.248f
#define KFOV    3.0f

// 5-tap normalized Gaussian for sigma = 0.248 (reference 2D kernel is exactly
// separable into this): computed offline, hardcoded so no per-thread expf.
#define G0 7.5380362e-15f
#define G1 2.9452601e-04f
#define G2 9.9941095e-01f

typedef float v2f  __attribute__((ext_vector_type(2)));
typedef float v8f  __attribute__((ext_vector_type(8)));
typedef unsigned int u32x4 __attribute__((ext_vector_type(4)));
typedef int   i32x4 __attribute__((ext_vector_type(4)));
typedef int   i32x8 __attribute__((ext_vector_type(8)));

__device__ __forceinline__ int iclampi(int v, int lo, int hi) {
    return v < lo ? lo : (v > hi ? hi : v);
}

// Fused (bilinear x2 upsample, align_corners=False) + 5-tap blur -> 2-phase
// 4-tap polyphase filter. Taps sit at input rows kb..kb+3 with
// kb = (outIdx>>1) - 2 + (outIdx&1).
__device__ __constant__ const float WVT[2][4] = {
    {0.25f * G0, 0.75f * G0 + 0.75f * G1 + 0.25f * G2,
     0.25f * G1 + 0.75f * G2 + 0.75f * G1 + 0.25f * G0,
     0.25f * G1 + 0.75f * G0},
    {0.75f * G0 + 0.25f * G1, 0.25f * G0 + 0.75f * G1 + 0.75f * G2 + 0.25f * G1,
     0.25f * G2 + 0.75f * G1 + 0.75f * G0, 0.25f * G0}};

// ---------------- eccentricity / blend weights (closed form) ----------------

__device__ __forceinline__ float eccR(const float* __restrict__ fx, float x, float y) {
    float dmin = 3.4e38f;
#pragma unroll
    for (int f = 0; f < 4; ++f) {
        float dx = x - fx[2 * f];
        float dy = y - fx[2 * f + 1];
        dmin = fminf(dmin, sqrtf(dx * dx + dy * dy));
    }
    float theta = dmin / 7.5f;
    return 2.5f / (theta + 2.5f);
}

template <int J>
__device__ __forceinline__ float tsC(float R) {
    if constexpr (J >= 5) {
        return 0.f;
    } else {
        constexpr float sc = 0.25f * (float)(1 << J);  // 2^(J-2)
        float e = (sc / SIGMA_F) * R;
        return expf(-(e * e) * KFOV);
    }
}

template <int J>
__device__ __forceinline__ float bsC(float R) {
    float tj = tsC<J>(R), tj1 = tsC<J + 1>(R);
    return (0.5f - tj1) / (tj - tj1 + 1e-5f);
}

__device__ __forceinline__ int layerOf(float R) {
    const float omg[6] = {0.47683023f, 0.23841511f, 0.11920756f,
                          0.05960378f, 0.02980189f, 0.f};
    int L = 0;
#pragma unroll
    for (int i = 1; i < 6; ++i) L = (R >= omg[i] && R <= omg[i - 1]) ? i : L;
    return L;
}

// Ms[P] at a pixel (compile-time level index => everything folds).
template <int P>
__device__ __forceinline__ float msWeightT(float R) {
    int L = layerOf(R);
    float w = 0.f;
    if constexpr (P == 0) {
        w = (L == 0) ? 1.f : w;
        w = (L == 1) ? bsC<0>(R) : w;
    } else if constexpr (P < 5) {
        w = (L == P) ? (1.f - bsC<P - 1>(R)) : w;
        w = (L == P + 1) ? bsC<P>(R) : w;
    } else {
        w = (L == 5) ? (1.f - bsC<4>(R)) : w;
    }
    return w;
}

// ---------------- TDM: async 2D fp32 tile load global -> LDS ----------------
__device__ __forceinline__ void tdm_load_tile(const float* __restrict__ src,
                                              unsigned lds_byte_addr,
                                              int rowStrideElems, int rows,
                                              int cols, int tensorW, int tensorH) {
    unsigned long long ga = (unsigned long long)(uintptr_t)src;
    u32x4 g0;
    g0.x = 1u;                                   // count=1, user mode, no gather
    g0.y = lds_byte_addr;                        // LDS byte address
    g0.z = (unsigned)(ga & 0xFFFFFFFFu);         // global_addr[31:0]
    g0.w = (unsigned)((ga >> 32) & 0x01FFFFFFu)  // global_addr[56:32]
           | (2u << 30);                         // type=2 ("image")
    i32x8 g1;
    g1[0] = (int)(2u << 16);                                         // data_size=4B, wg_mask=0
    g1[1] = (int)(((unsigned)tensorW & 0xFFFFu) << 16);              // tensor_dim0 lo16
    g1[2] = (int)((((unsigned)tensorW >> 16) & 0xFFFFu) |
                  (((unsigned)tensorH & 0xFFFFu) << 16));            // td0 hi / td1 lo
    g1[3] = (int)((((unsigned)tensorH >> 16) & 0xFFFFu) |
                  (((unsigned)cols & 0xFFFFu) << 16));               // td1 hi / tile_dim0
    g1[4] = (int)((unsigned)rows & 0xFFFFu);                         // tile_dim1 (tile_dim2=0)
    g1[5] = (int)(unsigned)rowStrideElems;                           // tensor_dim0_stride lo32
    g1[6] = 0;
    g1[7] = 0;
    i32x4 zz = (i32x4){0, 0, 0, 0};
#if defined(__clang_major__) && (__clang_major__ >= 23)
    i32x8 z8 = (i32x8){0, 0, 0, 0, 0, 0, 0, 0};
    __builtin_amdgcn_tensor_load_to_lds(g0, g1, zz, zz, z8, 0);
#else
    __builtin_amdgcn_tensor_load_to_lds(g0, g1, zz, zz, 0);
#endif
}

// --------------------------- pyramid kernels --------------------------------

// Fused zero-padded 5x5 blur + 2x2 mean pool: separable 6-tap stride-2.
__global__ void k_down(const float* __restrict__ in, float* __restrict__ out, int outN) {
    const int inN = outN * 2;
    int x = blockIdx.x * blockDim.x + threadIdx.x;
    int y = blockIdx.y * blockDim.y + threadIdx.y;
    int c = blockIdx.z;
    if (x >= outN || y >= outN) return;
    const float* ip = in + (size_t)c * inN * inN;

    const float dw[6] = {0.5f * G0, 0.5f * (G0 + G1), 0.5f * (G1 + G2),
                         0.5f * (G2 + G1), 0.5f * (G1 + G0), 0.5f * G0};

    float acc = 0.f;
#pragma unroll
    for (int t = 0; t < 6; ++t) {
        int m = 2 * y - 2 + t;
        if (m < 0 || m >= inN) continue;  // zero padding
        float row = 0.f;
#pragma unroll
        for (int s = 0; s < 6; ++s) {
            int mx = 2 * x - 2 + s;
            if (mx < 0 || mx >= inN) continue;
            row += dw[s] * ip[(size_t)m * inN + mx];
        }
        acc += dw[t] * row;
    }
    out[((size_t)c * outN + y) * outN + x] = acc;
}

// Fused bilinear x2 upsample (clamped sampling) + zero-padded 5x5 blur,
// n x n -> 2n x 2n (intermediate chain steps, sizes <= 1024).
__global__ void k_upblur(const float* __restrict__ in, float* __restrict__ out, int n) {
    const int on = n * 2;
    int x = blockIdx.x * blockDim.x + threadIdx.x;
    int y = blockIdx.y * blockDim.y + threadIdx.y;
    int c = blockIdx.z;
    if (x >= on || y >= on) return;
    const float* ip = in + (size_t)c * n * n;

    const float g[5] = {G0, G1, G2, G1, G0};
    float acc = 0.f;
#pragma unroll
    for (int ty = 0; ty < 5; ++ty) {
        int uy = y - 2 + ty;
        if (uy < 0 || uy >= on) continue;  // blur zero padding
        int ky = uy >> 1;
        int iy0, iy1;
        float ay0, ay1;
        if (uy & 1) { iy0 = ky; iy1 = (ky + 1 < n) ? ky + 1 : n - 1; ay0 = 0.75f; ay1 = 0.25f; }
        else        { iy0 = (ky - 1 >= 0) ? ky - 1 : 0; iy1 = ky;    ay0 = 0.25f; ay1 = 0.75f; }
#pragma unroll
        for (int tx = 0; tx < 5; ++tx) {
            int ux = x - 2 + tx;
            if (ux < 0 || ux >= on) continue;
            int kx = ux >> 1;
            int ix0, ix1;
            float ax0, ax1;
            if (ux & 1) { ix0 = kx; ix1 = (kx + 1 < n) ? kx + 1 : n - 1; ax0 = 0.75f; ax1 = 0.25f; }
            else        { ix0 = (kx - 1 >= 0) ? kx - 1 : 0; ix1 = kx;    ax0 = 0.25f; ax1 = 0.75f; }
            float v = ay0 * (ax0 * ip[(size_t)iy0 * n + ix0] + ax1 * ip[(size_t)iy0 * n + ix1]) +
                      ay1 * (ax0 * ip[(size_t)iy1 * n + ix0] + ax1 * ip[(size_t)iy1 * n + ix1]);
            acc += g[ty] * g[tx] * v;
        }
    }
    out[((size_t)c * on + y) * on + x] = acc;
}

// Level 0: out = Ms[0] * img  (initializes the accumulator)
__global__ void k_final0(const float* __restrict__ img, const float* __restrict__ fx,
                         float* __restrict__ out) {
    const int N = 2048;
    const size_t HW = (size_t)N * N;
    size_t idx = (size_t)blockIdx.x * blockDim.x + threadIdx.x;
    if (idx >= HW) return;
    int y = (int)(idx / N), x = (int)(idx % N);
    float m0 = msWeightT<0>(eccR(fx, (float)x, (float)y));
#pragma unroll
    for (int c = 0; c < 3; ++c) out[(size_t)c * HW + idx] = m0 * img[(size_t)c * HW + idx];
}

// Levels 1..5 final step: 1024^2 chain result -> fused (up x2 + blur) at full
// res + blend-accumulate. One wave per 16x16 output tile. TDM stages the
// 12x12x3 input tiles into LDS; vertical polyphase filter runs as a banded
// GEMM on V_WMMA_F32_16X16X4_F32 (3 K-chunks of 4).
template <int P>
__global__ __launch_bounds__(32) void k_final_wmma(const float* __restrict__ in,
                                                   const float* __restrict__ fx,
                                                   float* __restrict__ out) {
    const int inN = 1024;
    const int N = 2048;
    __shared__ float sIn[3][12 * 12];
    __shared__ float sT[12 * 16];
    __shared__ float sWv[12 * 16];  // banded vertical weights, [r][M]

    const int lane = (int)threadIdx.x;
    const int X0 = (int)blockIdx.x * 16;
    const int Y0 = (int)blockIdx.y * 16;
    const int gy0 = iclampi(Y0 / 2 - 2, 0, inN - 12);
    const int gx0 = iclampi(X0 / 2 - 2, 0, inN - 12);

    // Kick off the 3 channel tile loads (TENSORcnt pipeline, depth 3).
#pragma unroll
    for (int ch = 0; ch < 3; ++ch) {
        const float* src = in + (((size_t)ch * inN) + gy0) * inN + gx0;
        tdm_load_tile(src, (unsigned)(uintptr_t)&sIn[ch][0], inN, 12, 12, inN, inN);
    }

    const int half = lane >> 4;    // C/D layout: lanes 16-31 hold M+8
    const int nn = lane & 15;      // column (and A-row) index this lane owns

    // --- per-lane hoisted horizontal filter state (column nn, fixed) ---
    const int phx = nn & 1;
    const int kbx = ((X0 + nn) >> 1) - 2 + phx;
    int colx[4];
    float whl[4];
#pragma unroll
    for (int i = 0; i < 4; ++i) {
        colx[i] = iclampi(kbx + i, 0, inN - 1) - gx0;
        whl[i] = WVT[phx][i];
    }

    // --- banded vertical weight matrix Wv[16 x 12] -> LDS (once per block) ---
    {
        const int M = nn;
        const int phM = M & 1;
        const int kbM = ((Y0 + M) >> 1) - 2 + phM;
#pragma unroll
        for (int k = 0; k < 6; ++k) {
            int r = 2 * k + half;
            float w = 0.f;
#pragma unroll
            for (int j = 0; j < 4; ++j) {
                int cr = iclampi(kbM + j, 0, inN - 1) - gy0;
                w += (cr == r) ? WVT[phM][j] : 0.f;
            }
            sWv[r * 16 + M] = w;
        }
    }

    // Blend weights for the 8 output pixels this lane owns (C/D layout).
    float msv[8];
#pragma unroll
    for (int v = 0; v < 8; ++v) {
        float R = eccR(fx, (float)(X0 + nn), (float)(Y0 + v + half * 8));
        msv[v] = msWeightT<P>(R);
    }

#pragma unroll
    for (int ch = 0; ch < 3; ++ch) {
        if (ch == 0)      __builtin_amdgcn_s_wait_tensorcnt(2);
        else if (ch == 1) __builtin_amdgcn_s_wait_tensorcnt(1);
        else              __builtin_amdgcn_s_wait_tensorcnt(0);
        __syncthreads();

        // Horizontal polyphase pass: sT[12][16]; lane writes rows 2k+half.
#pragma unroll
        for (int k = 0; k < 6; ++k) {
            int r = 2 * k + half;
            const float* sr = &sIn[ch][r * 12];
            float acc = whl[0] * sr[colx[0]] + whl[1] * sr[colx[1]] +
                        whl[2] * sr[colx[2]] + whl[3] * sr[colx[3]];
            sT[r * 16 + nn] = acc;
        }
        __syncthreads();

        // Vertical polyphase as banded GEMM: D[16x16] = Wv[16x12] x sT[12x16].
        v8f d = {0.f, 0.f, 0.f, 0.f, 0.f, 0.f, 0.f, 0.f};
#pragma unroll
        for (int c4 = 0; c4 < 3; ++c4) {
            int k0 = 4 * c4 + 2 * half;  // A 16x4 f32: vgpr0->K=2*half, vgpr1->K=2*half+1
            v2f a, bf;
            a.x = sWv[(k0    ) * 16 + nn];
            a.y = sWv[(k0 + 1) * 16 + nn];
            bf.x = sT[(k0    ) * 16 + nn];  // B 4x16: vgpr v, lane-half h -> row v+2h
            bf.y = sT[(k0 + 1) * 16 + nn];
            d = __builtin_amdgcn_wmma_f32_16x16x4_f32(false, a, false, bf,
                                                      (short)0, d, false, false);
        }

        // Blend-accumulate into the output image (coalesced per row).
#pragma unroll
        for (int v = 0; v < 8; ++v) {
            int yy = Y0 + v + half * 8;
            size_t o = (((size_t)ch) * N + yy) * N + (X0 + nn);
            out[o] += msv[v] * d[v];
        }
        __syncthreads();
    }
}

// ------------------------------- launcher -----------------------------------

extern "C" void kernel_launch(void* const* d_in, const int* in_sizes, int n_in,
                              void* d_out, int out_size, void* d_ws, size_t ws_size,
                              hipStream_t stream) {
    (void)in_sizes; (void)n_in; (void)out_size; (void)ws_size;
    const float* img = (const float*)d_in[0];   // [3,2048,2048] fp32
    const float* fixs = (const float*)d_in[1];  // [4,2] fp32
    float* out = (float*)d_out;                 // [3,2048,2048] fp32
    float* ws = (float*)d_ws;                   // needs ~42 MB

    size_t off = 0;
    float* pyr1 = ws + off; off += (size_t)3 * 1024 * 1024;
    float* pyr2 = ws + off; off += (size_t)3 * 512 * 512;
    float* pyr3 = ws + off; off += (size_t)3 * 256 * 256;
    float* pyr4 = ws + off; off += (size_t)3 * 128 * 128;
    float* pyr5 = ws + off; off += (size_t)3 * 64 * 64;
    float* tmp0 = ws + off; off += (size_t)3 * 1024 * 1024;
    float* tmp1 = ws + off; off += (size_t)3 * 1024 * 1024;

    dim3 b16(16, 16);
    // Gaussian pyramid (fused blur + 2x2 pool)
    k_down<<<dim3(64, 64, 3), b16, 0, stream>>>(img,  pyr1, 1024);
    k_down<<<dim3(32, 32, 3), b16, 0, stream>>>(pyr1, pyr2, 512);
    k_down<<<dim3(16, 16, 3), b16, 0, stream>>>(pyr2, pyr3, 256);
    k_down<<<dim3(8,  8,  3), b16, 0, stream>>>(pyr3, pyr4, 128);
    k_down<<<dim3(4,  4,  3), b16, 0, stream>>>(pyr4, pyr5, 64);

    // Level 0 initializes the blend accumulator.
    k_final0<<<(2048 * 2048) / 256, 256, 0, stream>>>(img, fixs, out);

    // Levels 1..5: upsample chain to 1024, then fused final step + blend.
    const float* pyr[6] = {img, pyr1, pyr2, pyr3, pyr4, pyr5};
    for (int p = 1; p < 6; ++p) {
        const float* cur = pyr[p];
        int s = 2048 >> p;
        float* dst = tmp0;
        float* alt = tmp1;
        while (s < 1024) {
            int os = s * 2;
            k_upblur<<<dim3(os / 16, os / 16, 3), b16, 0, stream>>>(cur, dst, s);
            cur = dst;
            float* t = dst; dst = alt; alt = t;
            s = os;
        }
        dim3 gfin(128, 128);
        switch (p) {
            case 1: k_final_wmma<1><<<gfin, 32, 0, stream>>>(cur, fixs, out); break;
            case 2: k_final_wmma<2><<<gfin, 32, 0, stream>>>(cur, fixs, out); break;
            case 3: k_final_wmma<3><<<gfin, 32, 0, stream>>>(cur, fixs, out); break;
            case 4: k_final_wmma<4><<<gfin, 32, 0, stream>>>(cur, fixs, out); break;
            default: k_final_wmma<5><<<gfin, 32, 0, stream>>>(cur, fixs, out); break;
        }
    }
}